// GNN_21474836480432
// MI455X (gfx1250) — compile-verified
//
#include <hip/hip_runtime.h>
#include <hip/hip_bf16.h>
#include <stdint.h>

// GNN forward (eval mode), D = 96 fixed by reference.
// out = (X + sum_{l=1..3} ego_l) / 4,  ego_{l} = agg + agg*ego_{l-1},
// agg = COO-SpMM(vals, rows, cols, ego_{l-1}).
//
// Roofline: ~0.5 GFLOP total vs ~75 MB working set, all L2-resident (192 MB).
// Bound by L2 atomic-RMW + gather throughput -> use non-returning device-scope
// global_atomic_add_f32, async global->LDS staging of edge streams, and a
// wave32-native layout (D = 96 = 3 x 32 lanes per edge).

#define EMB 96
#define EW_BLOCK 256
#define SPMM_BLOCK 256
#define SPMM_TILE 512  // edges staged per block tile (512*(4+4+4) = 6 KB LDS)

// ---------------------------------------------------------------------------
// Device-scope, non-returning f32 atomic add (L2 atomic units, STOREcnt).
// No "memory" clobber: the only consumer of agg is the *next* kernel, and
// S_ENDPGM performs an implicit wait-idle; leaving it off lets the compiler
// keep many independent gathers/atomics in flight per wave.
// ---------------------------------------------------------------------------
__device__ __forceinline__ void atomic_add_f32_dev(float* p, float v) {
#if defined(__gfx1250__)
  asm volatile("global_atomic_add_f32 %0, %1, off scope:SCOPE_DEV"
               :
               : "v"((unsigned long long)(size_t)p), "v"(v));
#else
  atomicAdd(p, v);
#endif
}

// ---------------------------------------------------------------------------
// init: ego = X, out(acc) = X, agg = 0   (float4 over N*EMB)
// ---------------------------------------------------------------------------
__global__ void __launch_bounds__(EW_BLOCK) gnn_init(
    const float* __restrict__ X, float* __restrict__ ego,
    float* __restrict__ agg, float* __restrict__ out, int n4) {
  int i = blockIdx.x * blockDim.x + threadIdx.x;
  if (i < n4) {
    float4 x = ((const float4*)X)[i];
    ((float4*)ego)[i] = x;
    ((float4*)out)[i] = x;
    ((float4*)agg)[i] = make_float4(0.f, 0.f, 0.f, 0.f);
  }
}

// ---------------------------------------------------------------------------
// SpMM scatter: agg[rows[e], :] += vals[e] * ego[cols[e], :]
// Edge streams staged tile-wise into LDS with async global->LDS loads;
// one wave per edge, lane d covers dims {d, d+32, d+64}; edge loop
// unrolled x2 with prefetch of the following pair's source rows.
// ---------------------------------------------------------------------------
__global__ void __launch_bounds__(SPMM_BLOCK) gnn_spmm(
    const float* __restrict__ ego, const float* __restrict__ vals,
    const int* __restrict__ rows, const int* __restrict__ cols,
    float* __restrict__ agg, int nE) {
  __shared__ int s_rows[SPMM_TILE];
  __shared__ int s_cols[SPMM_TILE];
  __shared__ float s_vals[SPMM_TILE];

  const int lane = threadIdx.x & 31;
  const int wave = threadIdx.x >> 5;
  const int nWaves = SPMM_BLOCK / 32;  // 8

  for (int base = blockIdx.x * SPMM_TILE; base < nE;
       base += gridDim.x * SPMM_TILE) {
    const int cnt = min(SPMM_TILE, nE - base);

#if defined(__gfx1250__)
    if (cnt == SPMM_TILE) {
      // Each of 256 threads async-copies 8 bytes of each 2 KB edge array.
      const unsigned t8 = (unsigned)threadIdx.x * 8u;
      unsigned ldsR = (unsigned)(size_t)(&s_rows[0]) + t8;
      unsigned ldsC = (unsigned)(size_t)(&s_cols[0]) + t8;
      unsigned ldsV = (unsigned)(size_t)(&s_vals[0]) + t8;
      unsigned long long gR = (unsigned long long)(size_t)(rows + base) + t8;
      unsigned long long gC = (unsigned long long)(size_t)(cols + base) + t8;
      unsigned long long gV = (unsigned long long)(size_t)(vals + base) + t8;
      asm volatile("global_load_async_to_lds_b64 %0, %1, off"
                   :
                   : "v"(ldsR), "v"(gR)
                   : "memory");
      asm volatile("global_load_async_to_lds_b64 %0, %1, off"
                   :
                   : "v"(ldsC), "v"(gC)
                   : "memory");
      asm volatile("global_load_async_to_lds_b64 %0, %1, off"
                   :
                   : "v"(ldsV), "v"(gV)
                   : "memory");
      asm volatile("s_wait_asynccnt 0x0" ::: "memory");
    } else
#endif
    {
      for (int i = threadIdx.x; i < cnt; i += SPMM_BLOCK) {
        s_rows[i] = rows[base + i];
        s_cols[i] = cols[base + i];
        s_vals[i] = vals[base + i];
      }
    }
    __syncthreads();

    for (int j = wave; j < cnt; j += 2 * nWaves) {
      const int j1 = j + nWaves;

      // Prefetch the source rows of the *next* pair into WGP L0.
      const int jn0 = j + 2 * nWaves;
      const int jn1 = jn0 + nWaves;
      if (jn0 < cnt)
        __builtin_prefetch(ego + (size_t)s_cols[jn0] * EMB + lane, 0, 3);
      if (jn1 < cnt)
        __builtin_prefetch(ego + (size_t)s_cols[jn1] * EMB + lane, 0, 3);

      // Edge j
      const int r0 = s_rows[j];
      const int c0 = s_cols[j];
      const float v0 = s_vals[j];
      const float* __restrict__ s0 = ego + (size_t)c0 * EMB;
      float* d0 = agg + (size_t)r0 * EMB;
      const float a0 = v0 * s0[lane];
      const float a1 = v0 * s0[lane + 32];
      const float a2 = v0 * s0[lane + 64];

      if (j1 < cnt) {
        // Edge j+nWaves: issue its gathers before any atomics for MLP.
        const int r1 = s_rows[j1];
        const int c1 = s_cols[j1];
        const float v1 = s_vals[j1];
        const float* __restrict__ s1 = ego + (size_t)c1 * EMB;
        float* d1 = agg + (size_t)r1 * EMB;
        const float b0 = v1 * s1[lane];
        const float b1 = v1 * s1[lane + 32];
        const float b2 = v1 * s1[lane + 64];

        atomic_add_f32_dev(d0 + lane, a0);
        atomic_add_f32_dev(d0 + lane + 32, a1);
        atomic_add_f32_dev(d0 + lane + 64, a2);
        atomic_add_f32_dev(d1 + lane, b0);
        atomic_add_f32_dev(d1 + lane + 32, b1);
        atomic_add_f32_dev(d1 + lane + 64, b2);
      } else {
        atomic_add_f32_dev(d0 + lane, a0);
        atomic_add_f32_dev(d0 + lane + 32, a1);
        atomic_add_f32_dev(d0 + lane + 64, a2);
      }
    }
    __syncthreads();
  }
}

// ---------------------------------------------------------------------------
// update: ne = agg + agg*ego ; ego = ne ; out = (out + ne) * scale ; agg = 0
// scale = 1 for layers 0,1 ; scale = 0.25 fused final division on layer 2.
// ---------------------------------------------------------------------------
__global__ void __launch_bounds__(EW_BLOCK) gnn_update(
    float* __restrict__ agg, float* __restrict__ ego, float* __restrict__ out,
    float scale, int n4) {
  int i = blockIdx.x * blockDim.x + threadIdx.x;
  if (i < n4) {
    float4 a = ((float4*)agg)[i];
    float4 e = ((float4*)ego)[i];
    float4 ne;
    ne.x = fmaf(a.x, e.x, a.x);
    ne.y = fmaf(a.y, e.y, a.y);
    ne.z = fmaf(a.z, e.z, a.z);
    ne.w = fmaf(a.w, e.w, a.w);
    ((float4*)ego)[i] = ne;
    float4 o = ((float4*)out)[i];
    o.x = (o.x + ne.x) * scale;
    o.y = (o.y + ne.y) * scale;
    o.z = (o.z + ne.z) * scale;
    o.w = (o.w + ne.w) * scale;
    ((float4*)out)[i] = o;
    ((float4*)agg)[i] = make_float4(0.f, 0.f, 0.f, 0.f);
  }
}

// ---------------------------------------------------------------------------
extern "C" void kernel_launch(void* const* d_in, const int* in_sizes, int n_in,
                              void* d_out, int out_size, void* d_ws,
                              size_t ws_size, hipStream_t stream) {
  const float* X = (const float*)d_in[0];     // [N, 96]
  const float* vals = (const float*)d_in[1];  // [E]
  const int* rows = (const int*)d_in[2];      // [E]
  const int* cols = (const int*)d_in[3];      // [E]
  float* out = (float*)d_out;                 // [N, 96] f32

  const int nX = in_sizes[0];  // N * 96
  const int nE = in_sizes[1];  // E

  float* ego = (float*)d_ws;  // nX floats
  float* agg = ego + nX;      // nX floats

  const int n4 = nX / 4;  // EMB=96 -> divisible by 4
  const int ewGrid = (n4 + EW_BLOCK - 1) / EW_BLOCK;
  const int spGrid = (nE + SPMM_TILE - 1) / SPMM_TILE;

  gnn_init<<<ewGrid, EW_BLOCK, 0, stream>>>(X, ego, agg, out, n4);
  for (int l = 0; l < 3; ++l) {
    gnn_spmm<<<spGrid, SPMM_BLOCK, 0, stream>>>(ego, vals, rows, cols, agg,
                                                nE);
    gnn_update<<<ewGrid, EW_BLOCK, 0, stream>>>(
        agg, ego, out, (l == 2) ? 0.25f : 1.0f, n4);
  }
}